// RetinaFaceWrapper_90623809946077
// MI455X (gfx1250) — compile-verified
//
#include <hip/hip_runtime.h>
#include <cstdint>

typedef float v2f __attribute__((ext_vector_type(2)));
typedef float v8f __attribute__((ext_vector_type(8)));
typedef unsigned long long u64t;
typedef unsigned int u32t;

#define TOPK 4096
#define CAND_CAP (1 << 19)   /* 524288 candidate slots (expected ~275k) */
#define NBINS 2048
#define CONF_BITS_BASE 0x3F666666u  /* bits of 0.9f */

/* ---------------- K0: zero counters + histogram ---------------- */
__global__ void init_kernel(u32t* __restrict__ ctrs, u32t* __restrict__ hist) {
  int t = blockIdx.x * blockDim.x + threadIdx.x;
  if (t < 16) ctrs[t] = 0u;
  for (int i = t; i < NBINS; i += gridDim.x * blockDim.x) hist[i] = 0u;
}

/* ---------------- K1: stream conf, compact candidates + histogram ---------------- */
__global__ __launch_bounds__(1024)
void scan_kernel(const float* __restrict__ conf, int n,
                 u64t* __restrict__ cand, u32t* __restrict__ ctrs,
                 u32t* __restrict__ hist) {
  __shared__ u32t lh[NBINS];
  for (int i = threadIdx.x; i < NBINS; i += blockDim.x) lh[i] = 0u;
  __syncthreads();
  int stride = gridDim.x * blockDim.x;
  for (int i = blockIdx.x * blockDim.x + threadIdx.x; i < n; i += stride) {
    __builtin_prefetch(conf + 2 * (i + stride) + 1, 0, 0);  /* global_prefetch_b8 */
    float s = conf[2 * i + 1];
    if (s > 0.9f) {
      u32t bits = __float_as_uint(s);
      u32t bin = (bits - CONF_BITS_BASE) >> 10;
      if (bin > NBINS - 1u) bin = NBINS - 1u;
      atomicAdd(&lh[bin], 1u);
      u32t pos = atomicAdd(&ctrs[0], 1u);
      if (pos < (u32t)CAND_CAP)
        cand[pos] = ((u64t)bits << 32) | (u64t)(0xFFFFFFFFu - (u32t)i);
    }
  }
  __syncthreads();
  for (int i = threadIdx.x; i < NBINS; i += blockDim.x)
    if (lh[i]) atomicAdd(&hist[i], lh[i]);
}

/* ---------------- K2: pick boundary bin from histogram ---------------- */
__global__ void pickbin_kernel(u32t* __restrict__ ctrs, const u32t* __restrict__ hist) {
  if (threadIdx.x != 0) return;
  int total = 0;
  for (int b = 0; b < NBINS; ++b) total += (int)hist[b];
  int desired = total < TOPK ? total : TOPK;
  int bstar = -2, chi = 0, cum = 0;
  if (desired > 0) {
    for (int b = NBINS - 1; b >= 0; --b) {
      int h = (int)hist[b];
      if (cum + h >= desired) { bstar = b; chi = cum; break; }
      cum += h;
    }
  }
  ctrs[3] = (u32t)bstar;
  ctrs[4] = (u32t)chi;
  ctrs[5] = (u32t)desired;
}

/* ---------------- K3: partition candidates into hi-set and boundary bin ---------------- */
__global__ void partition_kernel(const u64t* __restrict__ cand, u32t* __restrict__ ctrs,
                                 u64t* __restrict__ sel, u64t* __restrict__ bnd) {
  int m = (int)ctrs[0]; if (m > CAND_CAP) m = CAND_CAP;
  int bstar = (int)ctrs[3];
  int i = blockIdx.x * blockDim.x + threadIdx.x;
  if (i >= m) return;
  u64t k = cand[i];
  u32t bits = (u32t)(k >> 32);
  u32t bin = (bits - CONF_BITS_BASE) >> 10;
  if (bin > NBINS - 1u) bin = NBINS - 1u;
  if ((int)bin > bstar) {
    u32t p = atomicAdd(&ctrs[1], 1u);
    if (p < (u32t)TOPK) sel[p] = k;
  } else if ((int)bin == bstar) {
    u32t p = atomicAdd(&ctrs[2], 1u);
    if (p < (u32t)TOPK) bnd[p] = k;
  }
}

/* ---------------- bitonic sort (descending) of 4096 u64 keys in LDS ---------------- */
__device__ inline void bitonic_desc_4096(u64t* s, int tid) {
  for (int k = 2; k <= TOPK; k <<= 1) {
    for (int j = k >> 1; j > 0; j >>= 1) {
      __syncthreads();
      for (int t = tid; t < TOPK; t += 1024) {
        int x = t ^ j;
        if (x > t) {
          u64t a = s[t], b = s[x];
          bool up = (t & k) == 0;
          bool sw = up ? (a < b) : (a > b);   /* descending overall */
          if (sw) { s[t] = b; s[x] = a; }
        }
      }
    }
  }
  __syncthreads();
}

/* ---------------- K4: sort boundary, fill selection, sort selection ---------------- */
__global__ __launch_bounds__(1024)
void sortsel_kernel(u64t* __restrict__ sel, const u64t* __restrict__ bnd,
                    const u32t* __restrict__ ctrs) {
  __shared__ u64t sk[TOPK];
  int tid = threadIdx.x;
  int desired = (int)ctrs[5];
  int chi = (int)ctrs[4];
  int bcnt = (int)ctrs[2]; if (bcnt > TOPK) bcnt = TOPK;
  /* phase A: sort boundary bin, append the needed top entries */
  for (int t = tid; t < TOPK; t += 1024) sk[t] = (t < bcnt) ? bnd[t] : 0ull;
  bitonic_desc_4096(sk, tid);
  int needed = desired - chi; if (needed < 0) needed = 0;
  for (int t = tid; t < TOPK; t += 1024) {
    if (t < needed) sel[chi + t] = sk[t];
    if (t >= desired) sel[t] = 0ull;  /* padding when < 4096 candidates */
  }
  __threadfence();
  __syncthreads();
  /* phase B: full sort of the 4096 selected keys */
  for (int t = tid; t < TOPK; t += 1024) sk[t] = sel[t];
  bitonic_desc_4096(sk, tid);
  for (int t = tid; t < TOPK; t += 1024) sel[t] = sk[t];
}

/* ---------------- K5: gather + decode boxes/landmarks, stage output ---------------- */
__global__ void decode_kernel(const u64t* __restrict__ sel,
                              const float* __restrict__ loc,
                              const float* __restrict__ landm,
                              const float* __restrict__ priors,
                              float* __restrict__ out,
                              float* __restrict__ bx,
                              float* __restrict__ sc) {
  int r = blockIdx.x * blockDim.x + threadIdx.x;
  if (r >= TOPK) return;
  u64t key = sel[r];
  u32t hb = (u32t)(key >> 32);
  bool valid = hb != 0u;
  u32t idx = valid ? (0xFFFFFFFFu - (u32t)key) : 0u;
  float score = __uint_as_float(hb);             /* 0.0f for padded rows */
  const float* p = priors + (size_t)idx * 4;
  const float* lo = loc + (size_t)idx * 4;
  float pcx = p[0], pcy = p[1], pw = p[2], ph = p[3];
  float cx = pcx + lo[0] * 0.1f * pw;
  float cy = pcy + lo[1] * 0.1f * ph;
  float bw = pw * expf(lo[2] * 0.2f);
  float bh = ph * expf(lo[3] * 0.2f);
  float x1 = (cx - 0.5f * bw) * 8192.0f;
  float y1 = (cy - 0.5f * bh) * 8192.0f;
  float x2 = (cx + 0.5f * bw) * 8192.0f;
  float y2 = (cy + 0.5f * bh) * 8192.0f;
  float* o = out + (size_t)r * 15;
  o[0] = x1; o[1] = y1; o[2] = x2; o[3] = y2; o[4] = score;
  const float* lm = landm + (size_t)idx * 10;
#pragma unroll
  for (int k = 0; k < 5; ++k) {
    o[5 + 2 * k]     = (pcx + lm[2 * k]     * 0.1f * pw) * 8192.0f;
    o[5 + 2 * k + 1] = (pcy + lm[2 * k + 1] * 0.1f * ph) * 8192.0f;
  }
  bx[(size_t)r * 4 + 0] = x1; bx[(size_t)r * 4 + 1] = y1;
  bx[(size_t)r * 4 + 2] = x2; bx[(size_t)r * 4 + 3] = y2;
  sc[r] = score;
}

/* ---------------- K6: 16x16 IoU tiles -> suppression bitmask (WMMA + async-LDS) ---------------- */
__global__ __launch_bounds__(32)
void iou_kernel(const float* __restrict__ bx, unsigned short* __restrict__ sup) {
  __shared__ float lb[128];                 /* [0..63] i-boxes, [64..127] j-boxes */
  int l = threadIdx.x;
  int i0 = blockIdx.y * 16, j0 = blockIdx.x * 16;
  const float* src = (l < 16) ? (bx + (size_t)(i0 + l) * 4)
                              : (bx + (size_t)(j0 + (l - 16)) * 4);
  u32t ldsoff = (u32t)(size_t)(&lb[l * 4]);
  u64t gaddr = (u64t)(size_t)src;
  /* CDNA5 async tensor path: global -> LDS, tracked on ASYNCcnt */
  asm volatile("global_load_async_to_lds_b128 %0, %1, off"
               :: "v"(ldsoff), "v"(gaddr) : "memory");
  asm volatile("s_wait_asynccnt 0" ::: "memory");
  __syncthreads();

  int h = l >> 4, q = l & 15;
  float jx1 = lb[64 + q * 4 + 0], jy1 = lb[64 + q * 4 + 1];
  float jx2 = lb[64 + q * 4 + 2], jy2 = lb[64 + q * 4 + 3];
  float qx1 = lb[q * 4 + 0], qy1 = lb[q * 4 + 1];
  float qx2 = lb[q * 4 + 2], qy2 = lb[q * 4 + 3];
  float areaI = fmaxf(qx2 - qx1, 0.0f) * fmaxf(qy2 - qy1, 0.0f);
  float areaJ = fmaxf(jx2 - jx1, 0.0f) * fmaxf(jy2 - jy1, 0.0f);

  /* D[i][j] = area_i + area_j via v_wmma_f32_16x16x4_f32:
     A (16x4): A[i][0]=area_i, A[i][1]=1  -> lanes 0-15 hold K0,K1
     B (4x16): B[0][j]=1, B[1][j]=area_j  -> lanes 0-15 K0, lanes 16-31 K1 */
  v2f a, b;
  a.x = (h == 0) ? areaI : 0.0f;
  a.y = (h == 0) ? 1.0f : 0.0f;
  b.x = (h == 0) ? 1.0f : areaJ;
  b.y = 0.0f;
  v8f c = {};
  v8f d = __builtin_amdgcn_wmma_f32_16x16x4_f32(false, a, false, b,
                                                (short)0, c, false, false);
#pragma unroll
  for (int v = 0; v < 8; ++v) {
    int il = v + 8 * h;
    float ix1 = lb[il * 4 + 0], iy1 = lb[il * 4 + 1];
    float ix2 = lb[il * 4 + 2], iy2 = lb[il * 4 + 3];
    float wx = fmaxf(fminf(ix2, jx2) - fmaxf(ix1, jx1), 0.0f);
    float wy = fmaxf(fminf(iy2, jy2) - fmaxf(iy1, jy1), 0.0f);
    float inter = wx * wy;
    int gi = i0 + il, gj = j0 + q;
    /* iou > 0.5  <=>  3*inter > (areaI+areaJ) + 1e-12 */
    bool pred = (3.0f * inter > d[v] + 1e-12f) && (gj > gi);
    u32t m = __builtin_amdgcn_ballot_w32(pred);
    if (l == 0)
      sup[(size_t)(i0 + v) * 256 + (j0 >> 4)] = (unsigned short)(m & 0xFFFFu);
    if (l == 16)
      sup[(size_t)(i0 + v + 8) * 256 + (j0 >> 4)] = (unsigned short)(m >> 16);
  }
}

/* ---------------- K7: sequential greedy NMS, one wave, keep bits in registers ---------------- */
__global__ __launch_bounds__(32)
void nms_kernel(const unsigned short* __restrict__ sup,
                const float* __restrict__ sc, u32t* __restrict__ keepw) {
  int l = threadIdx.x;
  u32t keep[4];
#pragma unroll
  for (int r = 0; r < 4; ++r) {
    int w = l + 32 * r;
    u32t m = 0;
    for (int bbit = 0; bbit < 32; ++bbit)
      m |= (sc[w * 32 + bbit] > 0.9f ? 1u : 0u) << bbit;
    keep[r] = m;
  }
  const u32t* sup32 = (const u32t*)sup;     /* row stride = 128 dwords */
  for (int i = 0; i < TOPK; ++i) {
    int w = i >> 5;
    int r = w >> 5;
    int srclane = w & 31;
    u32t kv = (r == 0) ? keep[0] : (r == 1) ? keep[1] : (r == 2) ? keep[2] : keep[3];
    u32t kw = (u32t)__builtin_amdgcn_ds_bpermute(srclane << 2, (int)kv);
    if ((kw >> (i & 31)) & 1u) {
      const u32t* row = sup32 + (size_t)i * 128;
      keep[0] &= ~row[l];
      keep[1] &= ~row[l + 32];
      keep[2] &= ~row[l + 64];
      keep[3] &= ~row[l + 96];
    }
  }
#pragma unroll
  for (int r = 0; r < 4; ++r) keepw[l + 32 * r] = keep[r];
}

/* ---------------- K8: apply keep mask to output ---------------- */
__global__ void finalize_kernel(float* __restrict__ out, const u32t* __restrict__ keepw) {
  int r = blockIdx.x * blockDim.x + threadIdx.x;
  if (r >= TOPK) return;
  float f = ((keepw[r >> 5] >> (r & 31)) & 1u) ? 1.0f : 0.0f;
  float* o = out + (size_t)r * 15;
#pragma unroll
  for (int cidx = 0; cidx < 15; ++cidx) o[cidx] *= f;
}

extern "C" void kernel_launch(void* const* d_in, const int* in_sizes, int n_in,
                              void* d_out, int out_size, void* d_ws, size_t ws_size,
                              hipStream_t stream) {
  const float* loc    = (const float*)d_in[0];
  const float* conf   = (const float*)d_in[1];
  const float* landm  = (const float*)d_in[2];
  const float* priors = (const float*)d_in[3];
  int N = in_sizes[0] / 4;
  float* out = (float*)d_out;

  char* w = (char*)d_ws;
  u32t* ctrs = (u32t*)(w);                       /*      64 B */
  u32t* hist = (u32t*)(w + 64);                  /*    8192 B */
  u64t* sel  = (u64t*)(w + 8448);                /*   32768 B */
  u64t* bnd  = (u64t*)(w + 8448 + 32768);        /*   32768 B -> 73984 */
  float* bxp = (float*)(w + 73984);              /*   65536 B -> 139520 */
  float* scp = (float*)(w + 139520);             /*   16384 B -> 155904 */
  u32t* keepw = (u32t*)(w + 155904);             /*     512 B -> 156416 */
  unsigned short* sup = (unsigned short*)(w + 156416); /* 2 MB -> 2253568 */
  u64t* cand = (u64t*)(w + 2253568);             /*    4 MB -> 6447872 */

  init_kernel<<<8, 256, 0, stream>>>(ctrs, hist);
  scan_kernel<<<512, 1024, 0, stream>>>(conf, N, cand, ctrs, hist);
  pickbin_kernel<<<1, 64, 0, stream>>>(ctrs, hist);
  partition_kernel<<<CAND_CAP / 256, 256, 0, stream>>>(cand, ctrs, sel, bnd);
  sortsel_kernel<<<1, 1024, 0, stream>>>(sel, bnd, ctrs);
  decode_kernel<<<TOPK / 128, 128, 0, stream>>>(sel, loc, landm, priors, out, bxp, scp);
  dim3 g(256, 256);
  iou_kernel<<<g, 32, 0, stream>>>(bxp, sup);
  nms_kernel<<<1, 32, 0, stream>>>(sup, scp, keepw);
  finalize_kernel<<<TOPK / 128, 128, 0, stream>>>(out, keepw);
}